// TacoFusion_52999896433211
// MI455X (gfx1250) — compile-verified
//
#include <hip/hip_runtime.h>
#include <hip/hip_bf16.h>
#include <math.h>

typedef __attribute__((ext_vector_type(2))) float v2f;
typedef __attribute__((ext_vector_type(8))) float v8f;

#define D_DIM 1024
#define K_ADP 8
#define BS_ROWS 32768           // B*S = 8*4096
#define SEQ 4096
#define TILE_K 128
#define INV_SQRT_D 0.03125f     // 1/sqrt(1024)

// ---------------------------------------------------------------------------
// K1: bfrag[c][lane][j] = M[4c + (lane>=16?2:0) + j][lane&15], M = Wq^T @ keys
//     (pre-swizzled into the V_WMMA_F32_16X16X4_F32 B-operand lane layout;
//      cols 8..15 zero-padded). bfrag has 256*32*2 = 16384 floats.
// ---------------------------------------------------------------------------
__global__ void prep_kernel(const float* __restrict__ Wq,
                            const float* __restrict__ keys,
                            float* __restrict__ bfrag) {
    int idx = blockIdx.x * blockDim.x + threadIdx.x;   // 0..16383
    if (idx >= 256 * 64) return;
    int c = idx >> 6;
    int l = (idx >> 1) & 31;
    int j = idx & 1;
    int d = 4 * c + ((l >= 16) ? 2 : 0) + j;           // K index into M
    int n = l & 15;                                    // N column (adapter)
    float acc = 0.0f;
    if (n < K_ADP) {
        for (int e = 0; e < D_DIM; ++e)
            acc += Wq[e * D_DIM + d] * keys[e * K_ADP + n];
    }
    bfrag[idx] = acc;
}

// ---------------------------------------------------------------------------
// K2: t[r][k] = X[r][:] . M[:][k]  via V_WMMA_F32_16X16X4_F32.
//     One wave (32 threads) per 16-row tile. X staged through LDS with
//     coalesced b128 loads; A fragments via lane-addressed ds b64 reads.
// ---------------------------------------------------------------------------
__global__ __launch_bounds__(32)
void gemm_kernel(const float* __restrict__ x,
                 const float* __restrict__ bfrag,
                 float* __restrict__ tbuf) {
    __shared__ float lds[16 * TILE_K];                 // 8 KB
    const int lane = threadIdx.x;                      // 0..31
    const size_t row0 = (size_t)blockIdx.x * 16;       // 16-row tile base
    const float* xrow = x + row0 * D_DIM;

    v8f c = {0.f, 0.f, 0.f, 0.f, 0.f, 0.f, 0.f, 0.f};
    const int arow = lane & 15;                        // A: M = lane (mod 16)
    const int koff = (lane >> 4) * 2;                  // A: K pair 0/1 or 2/3

    for (int chunk = 0; chunk < D_DIM / TILE_K; ++chunk) {
        // stage 16 x TILE_K slab of X into LDS (coalesced: 32 lanes x 16B = one 512B row chunk)
        const float* src = xrow + chunk * TILE_K;
        #pragma unroll
        for (int r = 0; r < 16; ++r) {
            float4 v = *(const float4*)(src + (size_t)r * D_DIM + lane * 4);
            *(float4*)(&lds[r * TILE_K + lane * 4]) = v;
        }
        __syncthreads();

        const float* bsrc = bfrag + (size_t)chunk * (TILE_K / 4) * 64;
        #pragma unroll
        for (int kk = 0; kk < TILE_K; kk += 4) {
            v2f a = *(const v2f*)(&lds[arow * TILE_K + kk + koff]);
            v2f b = *(const v2f*)(bsrc + (kk >> 2) * 64 + lane * 2);
            // D = A(16x4 f32) * B(4x16 f32) + C(16x16 f32)
            c = __builtin_amdgcn_wmma_f32_16x16x4_f32(
                    false, a, false, b, (short)0, c, false, false);
        }
        __syncthreads();
    }

    // C/D layout: col = lane&15, row = v + (lane<16 ? 0 : 8). Keep cols 0..7.
    const int col = lane & 15;
    if (col < K_ADP) {
        const int rbase = (lane >> 4) * 8;
        #pragma unroll
        for (int v = 0; v < 8; ++v)
            tbuf[(row0 + rbase + v) * K_ADP + col] = c[v];
    }
}

// ---------------------------------------------------------------------------
// K3: per (b,k): inclusive cumsum over s of t, scaled by 1/((s+1)*sqrt(D)),
//     plus (bq . keys[:,k])/sqrt(D). In-place over tbuf. One wave per block.
// ---------------------------------------------------------------------------
__global__ __launch_bounds__(32)
void scan_kernel(const float* __restrict__ bq,
                 const float* __restrict__ keys,
                 float* __restrict__ tbuf) {
    const int lane = threadIdx.x;
    const int b = blockIdx.x >> 3;
    const int k = blockIdx.x & 7;

    // ck = (bq . keys[:,k]) / sqrt(D), wave-parallel reduction
    float ck = 0.0f;
    for (int e = lane; e < D_DIM; e += 32)
        ck += bq[e] * keys[e * K_ADP + k];
    #pragma unroll
    for (int off = 16; off > 0; off >>= 1)
        ck += __shfl_xor(ck, off, 32);
    ck *= INV_SQRT_D;

    float* base = tbuf + (size_t)b * SEQ * K_ADP + k;
    const int s0 = lane * (SEQ / 32);                  // 128 elems per lane

    float local = 0.0f;
    for (int i = 0; i < SEQ / 32; ++i)
        local += base[(size_t)(s0 + i) * K_ADP];

    // wave inclusive scan -> exclusive prefix for this lane
    float v = local;
    #pragma unroll
    for (int off = 1; off < 32; off <<= 1) {
        float n = __shfl_up(v, off, 32);
        if (lane >= off) v += n;
    }
    float run = v - local;

    for (int i = 0; i < SEQ / 32; ++i) {
        int s = s0 + i;
        run += base[(size_t)s * K_ADP];
        base[(size_t)s * K_ADP] = run * (INV_SQRT_D / (float)(s + 1)) + ck;
    }
}

// ---------------------------------------------------------------------------
// K4: softmax over the 8 adapters for each of the 32768 (b,s) positions.
// ---------------------------------------------------------------------------
__global__ void softmax_kernel(const float* __restrict__ sbuf,
                               float* __restrict__ out) {
    int i = blockIdx.x * blockDim.x + threadIdx.x;
    if (i >= BS_ROWS) return;
    const float* p = sbuf + (size_t)i * K_ADP;
    float m = p[0];
    #pragma unroll
    for (int k = 1; k < K_ADP; ++k) m = fmaxf(m, p[k]);
    float e[K_ADP];
    float sum = 0.0f;
    #pragma unroll
    for (int k = 0; k < K_ADP; ++k) { e[k] = __expf(p[k] - m); sum += e[k]; }
    float inv = 1.0f / sum;
    #pragma unroll
    for (int k = 0; k < K_ADP; ++k) out[(size_t)i * K_ADP + k] = e[k] * inv;
}

extern "C" void kernel_launch(void* const* d_in, const int* in_sizes, int n_in,
                              void* d_out, int out_size, void* d_ws, size_t ws_size,
                              hipStream_t stream) {
    const float* x    = (const float*)d_in[0];   // [8,4096,1024]
    const float* Wq   = (const float*)d_in[1];   // [1024,1024]
    const float* bq   = (const float*)d_in[2];   // [1024]
    const float* keys = (const float*)d_in[3];   // [1024,8]
    float* out = (float*)d_out;                  // [8,4096,8]

    float* bfrag = (float*)d_ws;                          // 16384 floats (64 KB)
    float* tbuf  = bfrag + 256 * 64;                      // 262144 floats (1 MB)

    prep_kernel<<<64, 256, 0, stream>>>(Wq, keys, bfrag);
    gemm_kernel<<<BS_ROWS / 16, 32, 0, stream>>>(x, bfrag, tbuf);
    scan_kernel<<<64, 32, 0, stream>>>(bq, keys, tbuf);
    softmax_kernel<<<(BS_ROWS + 255) / 256, 256, 0, stream>>>(tbuf, out);
}